// Fusion_89017492177331
// MI455X (gfx1250) — compile-verified
//
#include <hip/hip_runtime.h>

typedef __attribute__((ext_vector_type(2))) float v2f;
typedef __attribute__((ext_vector_type(8))) float v8f;

constexpr int B  = 8;
constexpr int N  = 4096;
constexpr int F  = 256;
constexpr int CH = 16;          // row chunks for column-mean partial sums

// ---------- helpers: top-k comparators matching jax.lax.top_k tie-break ----------
__device__ __forceinline__ bool lessVI(float v, int i, float w, int j) {
  return (v < w) || (v == w && i < j);
}
__device__ __forceinline__ bool greaterVI(float v, int i, float w, int j) {
  return (v > w) || (v == w && i < j);
}

// ---------- stage 1a: partial column sums of ref_flow (B x CH x F) ----------
__global__ void col_partial_kernel(const float* __restrict__ X, float* __restrict__ P) {
  const int b = blockIdx.x;
  const int c = blockIdx.y;
  const int f = threadIdx.x;                        // 256 threads, one per feature
  const int rowsPerChunk = N / CH;                  // 256
  const float* base = X + ((size_t)b * N + (size_t)c * rowsPerChunk) * F + f;
  float acc = 0.0f;
  for (int n = 0; n < rowsPerChunk; ++n) acc += base[(size_t)n * F];
  P[((size_t)b * CH + c) * F + f] = acc;
}

// ---------- stage 1b: reduce partials -> s[b,f] = mean over rows ----------
__global__ void col_reduce_kernel(const float* __restrict__ P, float* __restrict__ s) {
  const int b = blockIdx.x;
  const int f = threadIdx.x;
  float acc = 0.0f;
#pragma unroll
  for (int c = 0; c < CH; ++c) acc += P[((size_t)b * CH + c) * F + f];
  s[b * F + f] = acc * (1.0f / (float)N);
}

// ---------- stage 2: out[b,row] = dot(X[b,row,:], vec[b,:]) via fp32 WMMA ----------
// One wave computes 16 rows. A = 16x4 row tile, B = vec chunk replicated over
// 16 columns, C accumulates over 64 chunks of K=4. D's columns are identical,
// so lanes 0 and 16 each hold 8 final dot products in c[0..7].
__global__ void dot_rows_wmma_kernel(const float* __restrict__ X,
                                     const float* __restrict__ vec,
                                     float* __restrict__ out) {
  const int lane = threadIdx.x & 31;
  const int wave = (blockIdx.x * blockDim.x + threadIdx.x) >> 5;
  const int groupsPerBatch = N / 16;                // 256
  const int b = wave / groupsPerBatch;
  const int rowBase = (wave % groupsPerBatch) * 16;

  const float* Xb = X + ((size_t)b * N + rowBase) * F;
  const float* vb = vec + b * F;

  // A-matrix 16x4 f32 layout (ISA 7.12.2): lane L -> M = L&15, K = (L>>4)*2 + {0,1}
  const int m  = lane & 15;
  const int kp = (lane >> 4) << 1;

  v8f c = {};
#pragma unroll 4
  for (int kc = 0; kc < F; kc += 4) {
    v2f a  = *(const v2f*)(Xb + (size_t)m * F + kc + kp);   // 8B-aligned (even offsets)
    v2f bv = *(const v2f*)(vb + kc + kp);                   // column-replicated B
    // 8 args: (neg_a, A, neg_b, B, c_mod, C, reuse_a, reuse_b)
    c = __builtin_amdgcn_wmma_f32_16x16x4_f32(false, a, false, bv, (short)0, c,
                                              false, false);
  }

  // C/D layout: c[r] holds M=r (lanes 0-15) and M=r+8 (lanes 16-31); all columns equal.
  if (m == 0) {
    const int mBase = (lane >> 4) << 3;                     // 0 or 8
    float* o = out + (size_t)b * N + rowBase + mBase;
#pragma unroll
    for (int r = 0; r < 8; ++r) o[r] = c[r];
  }
}

// ---------- stage 3: per-batch top-3 smallest (ascending, tie -> lower index) ----------
__global__ void top3_min_kernel(const float* __restrict__ sim, int* __restrict__ kidx) {
  const int b = blockIdx.x;
  const float* s = sim + (size_t)b * N;
  float v[3]  = {3.4e38f, 3.4e38f, 3.4e38f};
  int   id[3] = {0x7fffffff, 0x7fffffff, 0x7fffffff};
  for (int i = threadIdx.x; i < N; i += blockDim.x) {
    const float x = s[i];
    if (lessVI(x, i, v[0], id[0])) { v[2]=v[1]; id[2]=id[1]; v[1]=v[0]; id[1]=id[0]; v[0]=x; id[0]=i; }
    else if (lessVI(x, i, v[1], id[1])) { v[2]=v[1]; id[2]=id[1]; v[1]=x; id[1]=i; }
    else if (lessVI(x, i, v[2], id[2])) { v[2]=x; id[2]=i; }
  }
  __shared__ float sv[256 * 3];
  __shared__ int   si[256 * 3];
#pragma unroll
  for (int j = 0; j < 3; ++j) { sv[threadIdx.x*3+j] = v[j]; si[threadIdx.x*3+j] = id[j]; }
  __syncthreads();
  if (threadIdx.x == 0) {
    float bv[3]  = {3.4e38f, 3.4e38f, 3.4e38f};
    int   bi[3]  = {0x7fffffff, 0x7fffffff, 0x7fffffff};
    for (int t = 0; t < 256 * 3; ++t) {
      const float x = sv[t]; const int i = si[t];
      if (lessVI(x, i, bv[0], bi[0])) { bv[2]=bv[1]; bi[2]=bi[1]; bv[1]=bv[0]; bi[1]=bi[0]; bv[0]=x; bi[0]=i; }
      else if (lessVI(x, i, bv[1], bi[1])) { bv[2]=bv[1]; bi[2]=bi[1]; bv[1]=x; bi[1]=i; }
      else if (lessVI(x, i, bv[2], bi[2])) { bv[2]=x; bi[2]=i; }
    }
    kidx[b*3+0] = bi[0]; kidx[b*3+1] = bi[1]; kidx[b*3+2] = bi[2];
  }
}

// ---------- stage 4: vmean[b,:] = mean of the 3 selected flow rows ----------
__global__ void build_vmean_kernel(const float* __restrict__ ref_flow,
                                   const int* __restrict__ kidx,
                                   float* __restrict__ vmean) {
  const int b = blockIdx.x;
  const int f = threadIdx.x;
  const int i0 = kidx[b*3+0], i1 = kidx[b*3+1], i2 = kidx[b*3+2];
  const float* base = ref_flow + (size_t)b * N * F;
  vmean[b * F + f] = (base[(size_t)i0 * F + f] +
                      base[(size_t)i1 * F + f] +
                      base[(size_t)i2 * F + f]) * (1.0f / 3.0f);
}

// ---------- stage 5: per-batch top-2 largest (descending, tie -> lower index) ----------
__global__ void top2_max_kernel(const float* __restrict__ sim, int* __restrict__ kidx2) {
  const int b = blockIdx.x;
  const float* s = sim + (size_t)b * N;
  float v[2]  = {-3.4e38f, -3.4e38f};
  int   id[2] = {0x7fffffff, 0x7fffffff};
  for (int i = threadIdx.x; i < N; i += blockDim.x) {
    const float x = s[i];
    if (greaterVI(x, i, v[0], id[0])) { v[1]=v[0]; id[1]=id[0]; v[0]=x; id[0]=i; }
    else if (greaterVI(x, i, v[1], id[1])) { v[1]=x; id[1]=i; }
  }
  __shared__ float sv[256 * 2];
  __shared__ int   si[256 * 2];
#pragma unroll
  for (int j = 0; j < 2; ++j) { sv[threadIdx.x*2+j] = v[j]; si[threadIdx.x*2+j] = id[j]; }
  __syncthreads();
  if (threadIdx.x == 0) {
    float bv[2] = {-3.4e38f, -3.4e38f};
    int   bi[2] = {0x7fffffff, 0x7fffffff};
    for (int t = 0; t < 256 * 2; ++t) {
      const float x = sv[t]; const int i = si[t];
      if (greaterVI(x, i, bv[0], bi[0])) { bv[1]=bv[0]; bi[1]=bi[0]; bv[0]=x; bi[0]=i; }
      else if (greaterVI(x, i, bv[1], bi[1])) { bv[1]=x; bi[1]=i; }
    }
    kidx2[b*2+0] = bi[0]; kidx2[b*2+1] = bi[1];
  }
}

// ---------- stage 6: gather output rows: [3 ref_rgb rows | 2 sup_rgb rows] ----------
__global__ void gather_out_kernel(const float* __restrict__ ref_rgb,
                                  const float* __restrict__ sup_rgb,
                                  const int* __restrict__ kidx,
                                  const int* __restrict__ kidx2,
                                  float* __restrict__ out) {
  const int b = blockIdx.x / 5;
  const int j = blockIdx.x % 5;
  const int f = threadIdx.x;
  int row; const float* src;
  if (j < 3) { row = kidx[b*3+j];       src = ref_rgb; }
  else       { row = kidx2[b*2+(j-3)];  src = sup_rgb; }
  out[((size_t)b * 5 + j) * F + f] = src[((size_t)b * N + row) * F + f];
}

extern "C" void kernel_launch(void* const* d_in, const int* in_sizes, int n_in,
                              void* d_out, int out_size, void* d_ws, size_t ws_size,
                              hipStream_t stream) {
  const float* ref_rgb  = (const float*)d_in[0];
  const float* ref_flow = (const float*)d_in[1];
  const float* sup_rgb  = (const float*)d_in[2];
  const float* sup_flow = (const float*)d_in[3];
  float* out = (float*)d_out;

  // workspace layout (floats)
  float* ws      = (float*)d_ws;
  float* s_mean  = ws;                    // B*F          = 2048
  float* vmean   = ws + 2048;             // B*F          = 2048
  float* intra   = ws + 4096;             // B*N          = 32768
  float* inter   = ws + 36864;            // B*N          = 32768
  float* P       = ws + 69632;            // B*CH*F       = 32768
  int*   kidx    = (int*)(ws + 102400);   // B*3 (pad 32)
  int*   kidx2   = kidx + 32;             // B*2

  // 1) s[b] = mean over rows of ref_flow[b]
  col_partial_kernel<<<dim3(B, CH), F, 0, stream>>>(ref_flow, P);
  col_reduce_kernel<<<B, F, 0, stream>>>(P, s_mean);

  // 2) intra_sim[b,n] = ref_flow[b,n] . s[b]   (fp32 WMMA; 2048 waves / 8 per block)
  dot_rows_wmma_kernel<<<(B * (N / 16)) / 8, 256, 0, stream>>>(ref_flow, s_mean, intra);

  // 3) top-3 smallest per batch
  top3_min_kernel<<<B, 256, 0, stream>>>(intra, kidx);

  // 4) vmean[b] = mean of 3 selected ref_flow rows
  build_vmean_kernel<<<B, F, 0, stream>>>(ref_flow, kidx, vmean);

  // 5) inter_sim_mean[b,m] = sup_flow[b,m] . vmean[b]
  dot_rows_wmma_kernel<<<(B * (N / 16)) / 8, 256, 0, stream>>>(sup_flow, vmean, inter);

  // 6) top-2 largest per batch
  top2_max_kernel<<<B, 256, 0, stream>>>(inter, kidx2);

  // 7) gather fusion output (B x 5 x F)
  gather_out_kernel<<<B * 5, F, 0, stream>>>(ref_rgb, sup_rgb, kidx, kidx2, out);
}